// HeteroRelConv_2723009266269
// MI455X (gfx1250) — compile-verified
//
#include <hip/hip_runtime.h>
#include <hip/hip_bf16.h>
#include <math.h>

// ---------------------------------------------------------------------------
// HypergraphConv x3 + pooling head for MI455X (gfx1250, wave32).
// Dense GEMMs: LDS-staged tiles + V_WMMA_F32_16X16X4_F32 (fp32 matrix pipe),
// branch-free inner loop (zero-padding resolved at LDS-staging time).
// Irregular 2M-incidence gather/scatter passes: fp32 streaming + atomics
// (memory/atomic bound; feature tables are L2-resident).
// ---------------------------------------------------------------------------

typedef float v2f __attribute__((ext_vector_type(2)));
typedef float v8f __attribute__((ext_vector_type(8)));

#define LRELU_NEG 0.2f

// ---------------- WMMA fp32 GEMM: C[M x Nc] = A[M x K] @ B[K x Nc] (+bias) --
// Grid: one block per 16-row M-strip. Block = 128 threads = 4 waves.
// LDS: A-strip [16 x Kpad] and B [Kpad x Ncpad], zero-padded at staging, so
// the WMMA K-loop is unconditional ds_load + v_wmma (EXEC all-1 throughout).
// Wave32 operand layouts per CDNA5 ISA 7.12.2:
//   A 16x4 f32 (2 VGPRs): lanes 0-15 rows M=0..15 K={k,k+1}; lanes 16-31 K={k+2,k+3}
//   B 4x16 f32 (2 VGPRs): lanes 0-15 cols N=0..15 K={k,k+1}; lanes 16-31 K={k+2,k+3}
//   C 16x16 f32 (8 VGPRs): vgpr r -> row r (lanes 0-15) / row r+8 (lanes 16-31)
// ACT: 0 = none, 1 = softplus.
template <int ACT>
__global__ void wmma_gemm_f32(const float* __restrict__ A,
                              const float* __restrict__ B,
                              const float* __restrict__ bias,
                              float* __restrict__ C,
                              int M, int K, int Nc) {
  extern __shared__ float smem[];
  const int Kpad  = (K + 3) & ~3;
  const int tiles_n = (Nc + 15) >> 4;
  const int Ncpad = tiles_n << 4;
  float* ldsA = smem;                  // [16 x Kpad]
  float* ldsB = smem + 16 * Kpad;      // [Kpad x Ncpad]

  const int m0 = blockIdx.x << 4;

  // ---- cooperative staging with zero-padding ----
  for (int idx = threadIdx.x; idx < 16 * Kpad; idx += blockDim.x) {
    const int row = idx / Kpad, kk = idx - row * Kpad;
    const int gr = m0 + row;
    ldsA[idx] = (gr < M && kk < K) ? A[(long)gr * K + kk] : 0.0f;
  }
  for (int idx = threadIdx.x; idx < Kpad * Ncpad; idx += blockDim.x) {
    const int kk = idx / Ncpad, col = idx - kk * Ncpad;
    ldsB[idx] = (kk < K && col < Nc) ? B[(long)kk * Nc + col] : 0.0f;
  }
  __syncthreads();

  const int lane = threadIdx.x & 31;
  const int wave = threadIdx.x >> 5;
  const int r    = lane & 15;           // row (A) / col (B) within tile
  const int koff = (lane >> 4) << 1;    // 0 or 2
  const int hi8  = (lane >> 4) << 3;    // 0 or 8 (C row offset)

  for (int nt = wave; nt < tiles_n; nt += (int)(blockDim.x >> 5)) {
    const int n0 = nt << 4;
    v8f acc = {};
    const float* a_row = ldsA + r * Kpad;
    const float* b_col = ldsB + n0 + r;
    for (int k = 0; k < Kpad; k += 4) {
      const int k0 = k + koff;
      v2f a, b;
      a.x = a_row[k0];
      a.y = a_row[k0 + 1];
      b.x = b_col[(long)k0 * Ncpad];
      b.y = b_col[(long)(k0 + 1) * Ncpad];
      acc = __builtin_amdgcn_wmma_f32_16x16x4_f32(false, a, false, b,
                                                  (short)0, acc, false, false);
    }
    const int cw = n0 + r;
    if (cw < Nc) {
      const float bv = bias ? bias[cw] : 0.0f;
#pragma unroll
      for (int i = 0; i < 8; ++i) {
        const int rw = m0 + hi8 + i;
        if (rw < M) {
          float v = acc[i] + bv;
          if (ACT == 1) v = (v > 20.0f) ? v : logf(1.0f + expf(v));
          C[(long)rw * Nc + cw] = v;
        }
      }
    }
  }
}

// ---------------- fills ----------------------------------------------------
__global__ void fill_f32(float* p, float v, long n) {
  long t = (long)blockIdx.x * blockDim.x + threadIdx.x;
  if (t < n) p[t] = v;
}
__global__ void fill_u32(unsigned* p, unsigned v, long n) {
  long t = (long)blockIdx.x * blockDim.x + threadIdx.x;
  if (t < n) p[t] = v;
}

// ---------------- monotone float<->uint key for atomicMax ------------------
__device__ __forceinline__ unsigned enc_f32(float f) {
  unsigned b = __float_as_uint(f);
  return (b & 0x80000000u) ? ~b : (b ^ 0x80000000u);
}
__device__ __forceinline__ float dec_f32(unsigned k) {
  unsigned b = (k & 0x80000000u) ? (k ^ 0x80000000u) : ~k;
  return __uint_as_float(b);
}
#define ENC_NEG_INF 0x007FFFFFu  // enc_f32(-inf)

// ---------------- wave-per-row dot with att half-vector ---------------------
// Coalesced: 32 lanes sweep consecutive channels, then shfl_xor reduction.
__global__ void rowdot_wave(const float* __restrict__ X,
                            const float* __restrict__ w,
                            float* __restrict__ out, int n, int H) {
  const int gid  = blockIdx.x * blockDim.x + threadIdx.x;
  const int wid  = gid >> 5;
  const int lane = threadIdx.x & 31;
  if (wid >= n) return;
  const float* row = X + (long)wid * H;
  float s = 0.0f;
  for (int c = lane; c < H; c += 32) s += row[c] * w[c];
#pragma unroll
  for (int off = 16; off > 0; off >>= 1) s += __shfl_xor(s, off, 32);
  if (lane == 0) out[wid] = s;
}

// ---------------- degrees (index-only, computed once) -----------------------
__global__ void degree_kernel(const int* __restrict__ node_idx,
                              const int* __restrict__ edge_idx,
                              float* __restrict__ Dcnt, float* __restrict__ Bcnt,
                              long NI) {
  long t = (long)blockIdx.x * blockDim.x + threadIdx.x;
  if (t >= NI) return;
  atomicAdd(&Dcnt[node_idx[t]], 1.0f);
  atomicAdd(&Bcnt[edge_idx[t]], 1.0f);
}
__global__ void invert_kernel(float* p, long n) {
  long t = (long)blockIdx.x * blockDim.x + threadIdx.x;
  if (t < n) { float v = p[t]; p[t] = (v > 0.0f) ? (1.0f / v) : 0.0f; }
}

// ---------------- attention score + segment max -----------------------------
__global__ void score_kernel(const float* __restrict__ s_n,
                             const float* __restrict__ s_e,
                             const int* __restrict__ node_idx,
                             const int* __restrict__ edge_idx,
                             float* __restrict__ raw,
                             unsigned* __restrict__ segmax, long NI) {
  long t = (long)blockIdx.x * blockDim.x + threadIdx.x;
  if (t >= NI) return;
  float v = s_n[node_idx[t]] + s_e[edge_idx[t]];
  v = (v > 0.0f) ? v : LRELU_NEG * v;           // leaky_relu
  raw[t] = v;
  atomicMax(&segmax[edge_idx[t]], enc_f32(v));
}

// ---------------- exp(score - m) + segment sum ------------------------------
__global__ void exp_kernel(float* __restrict__ alpha,               // in: raw, out: ex
                           const unsigned* __restrict__ segmax,
                           const int* __restrict__ edge_idx,
                           float* __restrict__ segsum, long NI) {
  long t = (long)blockIdx.x * blockDim.x + threadIdx.x;
  if (t >= NI) return;
  int e = edge_idx[t];
  float m = dec_f32(segmax[e]);
  if (!isfinite(m)) m = 0.0f;                   // PyG: non-finite max -> 0
  float ex = expf(alpha[t] - m);
  alpha[t] = ex;
  atomicAdd(&segsum[e], ex);
}

__global__ void norm_kernel(float* __restrict__ alpha,
                            const float* __restrict__ segsum,
                            const int* __restrict__ edge_idx, long NI) {
  long t = (long)blockIdx.x * blockDim.x + threadIdx.x;
  if (t >= NI) return;
  alpha[t] = alpha[t] / (segsum[edge_idx[t]] + 1e-16f);
}

// ---------------- pass 1: node -> hyperedge ---------------------------------
// thread t handles (incidence i, channel c); consecutive threads hit
// consecutive channels of one feature row -> coalesced gather + atomics.
__global__ void scatter_n2e(const float* __restrict__ xl,
                            const float* __restrict__ alpha,
                            const float* __restrict__ Binv,
                            const int* __restrict__ node_idx,
                            const int* __restrict__ edge_idx,
                            float* __restrict__ e_out, long NI, int H) {
  long t = (long)blockIdx.x * blockDim.x + threadIdx.x;
  if (t >= NI * H) return;
  long i = t / H; int c = (int)(t - i * H);
  int v = node_idx[i], e = edge_idx[i];
  float w = Binv[e] * alpha[i];
  atomicAdd(&e_out[(long)e * H + c], w * xl[(long)v * H + c]);
}

// ---------------- pass 2: hyperedge -> node ---------------------------------
__global__ void scatter_e2n(const float* __restrict__ e_out,
                            const float* __restrict__ alpha,
                            const float* __restrict__ Dinv,
                            const int* __restrict__ node_idx,
                            const int* __restrict__ edge_idx,
                            float* __restrict__ acc, long NI, int H) {
  long t = (long)blockIdx.x * blockDim.x + threadIdx.x;
  if (t >= NI * H) return;
  long i = t / H; int c = (int)(t - i * H);
  int v = node_idx[i], e = edge_idx[i];
  float w = Dinv[v] * alpha[i];
  atomicAdd(&acc[(long)v * H + c], w * e_out[(long)e * H + c]);
}

__global__ void add_bias_kernel(const float* __restrict__ acc,
                                const float* __restrict__ b,
                                float* __restrict__ h, long n, int H) {
  long t = (long)blockIdx.x * blockDim.x + threadIdx.x;
  if (t < n) h[t] = acc[t] + b[t % H];
}

// ---------------- scatter-mean pooling --------------------------------------
__global__ void pool_kernel(const float* __restrict__ h,
                            const int* __restrict__ batch,
                            float* __restrict__ pooled, float* __restrict__ cnt,
                            long n, int H) {
  long t = (long)blockIdx.x * blockDim.x + threadIdx.x;
  if (t >= n) return;
  long v = t / H; int c = (int)(t - v * H);
  int g = batch[v];
  atomicAdd(&pooled[(long)g * H + c], h[t]);
  if (c == 0) atomicAdd(&cnt[g], 1.0f);
}
__global__ void pool_div_kernel(float* __restrict__ pooled,
                                const float* __restrict__ cnt, long n, int H) {
  long t = (long)blockIdx.x * blockDim.x + threadIdx.x;
  if (t < n) pooled[t] /= fmaxf(cnt[t / H], 1.0f);
}

// ---------------- final linear [G,HOUT] @ [HOUT,1] --------------------------
__global__ void out_kernel(const float* __restrict__ z,
                           const float* __restrict__ out_W,
                           const float* __restrict__ out_b,
                           float* __restrict__ out, int G, int HOUT) {
  int g = blockIdx.x * blockDim.x + threadIdx.x;
  if (g >= G) return;
  float s = out_b[0];
  for (int k = 0; k < HOUT; ++k) s += z[(long)g * HOUT + k] * out_W[k];
  out[g] = s;
}

// ---------------------------------------------------------------------------
static inline int gdiv(long n, int b) { return (int)((n + b - 1) / b); }
static inline size_t gemm_lds_bytes(int K, int Nc) {
  int Kpad = (K + 3) & ~3;
  int Ncpad = ((Nc + 15) >> 4) << 4;
  return (size_t)(16 * Kpad + Kpad * Ncpad) * sizeof(float);
}

extern "C" void kernel_launch(void* const* d_in, const int* in_sizes, int n_in,
                              void* d_out, int out_size, void* d_ws, size_t ws_size,
                              hipStream_t stream) {
  const float* x        = (const float*)d_in[0];
  const float* he_attr  = (const float*)d_in[1];
  const int*   node_idx = (const int*)d_in[2];
  const int*   edge_idx = (const int*)d_in[3];
  const int*   batch    = (const int*)d_in[4];
  const float* embed_W  = (const float*)d_in[5];
  const float* embed_b  = (const float*)d_in[6];
  const float* lin_W    = (const float*)d_in[7];
  const float* att      = (const float*)d_in[8];
  const float* conv_b   = (const float*)d_in[9];
  const float* proj_W   = (const float*)d_in[10];
  const float* proj_b   = (const float*)d_in[11];
  const float* out_W    = (const float*)d_in[12];
  const float* out_b    = (const float*)d_in[13];
  float* out = (float*)d_out;

  const int H    = in_sizes[6];                 // 35
  const int F_IN = in_sizes[5] / H;             // 92
  const int HOUT = in_sizes[11];                // 64
  const int N    = in_sizes[0] / F_IN;          // 100000
  const int E    = in_sizes[1] / H;             // 50000
  const long NI  = in_sizes[2];                 // 2000000
  const int G    = out_size;                    // 256
  const int L    = in_sizes[8] / (2 * H);       // 3

  // -------- workspace carving (fp32 slots, 256B aligned) --------
  char* base = (char*)d_ws;
  size_t off = 0;
  auto carve = [&](long nelem) -> float* {
    float* p = (float*)(base + off);
    off += ((size_t)nelem * 4 + 255) & ~(size_t)255;
    return p;
  };
  float*    h      = carve((long)N * H);
  float*    xl     = carve((long)N * H);
  float*    el     = carve((long)E * H);
  float*    s_n    = carve(N);
  float*    s_e    = carve(E);
  float*    alpha  = carve(NI);
  unsigned* segmax = (unsigned*)carve(E);
  float*    segsum = carve(E);
  float*    Dinv   = carve(N);
  float*    Binv   = carve(E);
  float*    e_out  = carve((long)E * H);
  float*    acc    = carve((long)N * H);
  float*    pooled = carve((long)G * H);
  float*    cnt    = carve(G);
  float*    z      = carve((long)G * HOUT);
  (void)ws_size;

  const int T = 256;
  const long NH = (long)N * H, EH = (long)E * H, NIH = NI * H;

  // -------- embed: h = x @ embed_W + embed_b (WMMA fp32, LDS-staged) --------
  wmma_gemm_f32<0><<<gdiv(N, 16), 128, gemm_lds_bytes(F_IN, H), stream>>>(
      x, embed_W, embed_b, h, N, F_IN, H);

  // -------- degrees (index-only; once) --------
  fill_f32<<<gdiv(N, T), T, 0, stream>>>(Dinv, 0.0f, N);
  fill_f32<<<gdiv(E, T), T, 0, stream>>>(Binv, 0.0f, E);
  degree_kernel<<<gdiv(NI, T), T, 0, stream>>>(node_idx, edge_idx, Dinv, Binv, NI);
  invert_kernel<<<gdiv(N, T), T, 0, stream>>>(Dinv, N);
  invert_kernel<<<gdiv(E, T), T, 0, stream>>>(Binv, E);

  // -------- L hypergraph-attention layers --------
  for (int l = 0; l < L; ++l) {
    const float* W   = lin_W + (long)l * H * H;
    const float* a_n = att + (long)l * 2 * H;
    const float* a_e = a_n + H;
    const float* cb  = conv_b + (long)l * H;

    const size_t lds_hh = gemm_lds_bytes(H, H);
    wmma_gemm_f32<0><<<gdiv(N, 16), 128, lds_hh, stream>>>(h, W, nullptr, xl, N, H, H);
    wmma_gemm_f32<0><<<gdiv(E, 16), 128, lds_hh, stream>>>(he_attr, W, nullptr, el, E, H, H);

    rowdot_wave<<<gdiv((long)N * 32, T), T, 0, stream>>>(xl, a_n, s_n, N, H);
    rowdot_wave<<<gdiv((long)E * 32, T), T, 0, stream>>>(el, a_e, s_e, E, H);

    fill_u32<<<gdiv(E, T), T, 0, stream>>>(segmax, ENC_NEG_INF, E);
    fill_f32<<<gdiv(E, T), T, 0, stream>>>(segsum, 0.0f, E);

    score_kernel<<<gdiv(NI, T), T, 0, stream>>>(s_n, s_e, node_idx, edge_idx, alpha, segmax, NI);
    exp_kernel<<<gdiv(NI, T), T, 0, stream>>>(alpha, segmax, edge_idx, segsum, NI);
    norm_kernel<<<gdiv(NI, T), T, 0, stream>>>(alpha, segsum, edge_idx, NI);

    fill_f32<<<gdiv(EH, T), T, 0, stream>>>(e_out, 0.0f, EH);
    scatter_n2e<<<gdiv(NIH, T), T, 0, stream>>>(xl, alpha, Binv, node_idx, edge_idx, e_out, NI, H);

    fill_f32<<<gdiv(NH, T), T, 0, stream>>>(acc, 0.0f, NH);
    scatter_e2n<<<gdiv(NIH, T), T, 0, stream>>>(e_out, alpha, Dinv, node_idx, edge_idx, acc, NI, H);

    add_bias_kernel<<<gdiv(NH, T), T, 0, stream>>>(acc, cb, h, NH, H);
  }

  // -------- scatter-mean pooling --------
  fill_f32<<<gdiv((long)G * H, T), T, 0, stream>>>(pooled, 0.0f, (long)G * H);
  fill_f32<<<gdiv(G, T), T, 0, stream>>>(cnt, 0.0f, G);
  pool_kernel<<<gdiv(NH, T), T, 0, stream>>>(h, batch, pooled, cnt, NH, H);
  pool_div_kernel<<<gdiv((long)G * H, T), T, 0, stream>>>(pooled, cnt, (long)G * H, H);

  // -------- head: z = softplus(pooled @ proj_W + proj_b) (WMMA fp32) --------
  wmma_gemm_f32<1><<<gdiv(G, 16), 128, gemm_lds_bytes(H, HOUT), stream>>>(
      pooled, proj_W, proj_b, z, G, H, HOUT);

  // -------- out = z @ out_W + out_b --------
  out_kernel<<<gdiv(G, T), T, 0, stream>>>(z, out_W, out_b, out, G, HOUT);
}